// TokenTreeModel_68513318306334
// MI455X (gfx1250) — compile-verified
//
#include <hip/hip_runtime.h>

// D = A(16x4) x B(4x16) + C, f32 WMMA on gfx1250.
typedef __attribute__((ext_vector_type(2))) float v2f;
typedef __attribute__((ext_vector_type(8))) float v8f;

// Problem dims (from reference).
constexpr int   BB      = 2;
constexpr int   TT      = 512;
constexpr int   DEPTH   = 8;
constexpr int   VOCAB   = 32000;
constexpr int   N_BT    = BB * TT;              // 1024 (b,t) rows
constexpr int   CHUNK   = 64;                   // outputs per wave-iteration (4 tiles of 16)
constexpr int   CH_ROW  = VOCAB / CHUNK;        // 500 chunks per (b,t) row
constexpr long  NCHUNKS = (long)N_BT * CH_ROW;  // 512000 total chunks

__global__ __launch_bounds__(256) void
tokentree_depth_dot_wmma(const float* __restrict__ x,
                         const float* __restrict__ w,
                         const float* __restrict__ bias,
                         float* __restrict__ out) {
  const int lane = threadIdx.x & 31;
  const int half = lane >> 4;        // 0: K=0/1 (d low pair), 1: K=2/3
  const int col  = lane & 15;        // matrix column (v offset within tile)

  const int wave   = blockIdx.x * (blockDim.x >> 5) + (threadIdx.x >> 5);
  const int nwaves = gridDim.x * (blockDim.x >> 5);

  // Uniform scalar loads of the 8 weights + bias.
  const float w0 = w[0], w1 = w[1], w2 = w[2], w3 = w[3];
  const float w4 = w[4], w5 = w[5], w6 = w[6], w7 = w[7];
  const float bv = bias[0];

  // A-matrix (16x4, w broadcast down the M rows), native VGPR layout:
  // VGPR0 = K0 (lanes 0-15) / K2 (lanes 16-31); VGPR1 = K1 / K3.
  v2f a_lo, a_hi;
  a_lo.x = half ? w2 : w0;  a_lo.y = half ? w3 : w1;   // K = 0..3
  a_hi.x = half ? w6 : w4;  a_hi.y = half ? w7 : w5;   // K = 4..7

  const v8f cbias = {bv, bv, bv, bv, bv, bv, bv, bv};

  for (long c = wave; c < NCHUNKS; c += nwaves) {
    const long bt = c / CH_ROW;
    const int  v0 = (int)(c % CH_ROW) * CHUNK;

    const float* __restrict__ xp = x + bt * (long)(DEPTH * VOCAB);
    float* __restrict__       op = out + bt * (long)VOCAB + v0;

    float r[4];
#pragma unroll
    for (int t = 0; t < 4; ++t) {
      // B-matrix (4x16): lane n<16 -> column n, K=0,1 ; lanes 16-31 -> K=2,3.
      // d stride in memory is VOCAB floats, so these are b32 loads with
      // 64-byte coalesced half-wave segments.
      const float* base = xp + (long)(2 * half) * VOCAB + (v0 + 16 * t + col);
      v2f b_lo, b_hi;
      b_lo.x = base[0];             // d = 2*half
      b_lo.y = base[VOCAB];         // d = 2*half + 1
      b_hi.x = base[4 * VOCAB];     // d = 4 + 2*half
      b_hi.y = base[5 * VOCAB];     // d = 5 + 2*half

      // out = (w[4:8] . x) + ((w[0:4] . x) + bias), every D element of a lane
      // equals out[v0 + 16*t + (lane&15)].
      v8f d0 = __builtin_amdgcn_wmma_f32_16x16x4_f32(
          false, a_lo, false, b_lo, (short)0, cbias, false, false);
      v8f d1 = __builtin_amdgcn_wmma_f32_16x16x4_f32(
          false, a_hi, false, b_hi, (short)0, d0, false, false);
      r[t] = d1[0];
    }

    // Merge tile pairs into full-wave contiguous 128B stores:
    // lanes 0-15 carry tile 2k (cols 0-15), lanes 16-31 carry tile 2k+1.
    const float s0 = half ? r[1] : r[0];
    const float s1 = half ? r[3] : r[2];
    op[lane]      = s0;   // out[v0      + lane]
    op[32 + lane] = s1;   // out[v0 + 32 + lane]
  }
}

extern "C" void kernel_launch(void* const* d_in, const int* in_sizes, int n_in,
                              void* d_out, int out_size, void* d_ws, size_t ws_size,
                              hipStream_t stream) {
  (void)in_sizes; (void)n_in; (void)d_ws; (void)ws_size; (void)out_size;
  const float* x    = (const float*)d_in[0];
  const float* w    = (const float*)d_in[1];
  const float* bias = (const float*)d_in[2];
  float*       out  = (float*)d_out;

  // 4000 blocks x 8 waves = 32000 waves; 512000 chunks -> exactly 16 iters/wave.
  dim3 grid(4000), block(256);
  hipLaunchKernelGGL(tokentree_depth_dot_wmma, grid, block, 0, stream,
                     x, w, bias, out);
}